// CycleEmbedding_17368847745439
// MI455X (gfx1250) — compile-verified
//
#include <hip/hip_runtime.h>

typedef __attribute__((ext_vector_type(2))) float v2f;
typedef __attribute__((ext_vector_type(8))) float v8f;

#define NUM_SEG 100000   // cycle slots (M)
#define K_DIM   22       // atom codes (K)
#define K_PAD   24       // K padded to multiple of 4 for 16x16x4 WMMA
#define N_DIM   128      // hidden dim (N)
#define M_TILES (NUM_SEG / 16)        // 6250, exact
#define TILES_PER_BLOCK 5            // 6250 = 5 * 1250
#define NUM_BLOCKS (M_TILES / TILES_PER_BLOCK)

// ---------------------------------------------------------------------------
// Kernel 1: zero the count matrix in workspace (fp32 counts, exact integers).
// ---------------------------------------------------------------------------
__global__ void zero_counts_kernel(float* __restrict__ cnt, int n) {
    int i = blockIdx.x * blockDim.x + threadIdx.x;
    if (i < n) cnt[i] = 0.0f;
}

// ---------------------------------------------------------------------------
// Kernel 2: build Count[c, k] += 1 for each edge e:
//   atom = atom_to_cycle[0][e], cyc = atom_to_cycle[1][e], k = x[atom].
// 600K scalar fp32 atomics spread over 2.4M locations -> low contention.
// ---------------------------------------------------------------------------
__global__ void build_counts_kernel(const int* __restrict__ x,
                                    const int* __restrict__ a2c,
                                    float* __restrict__ cnt, int E) {
    int e = blockIdx.x * blockDim.x + threadIdx.x;
    if (e >= E) return;
    int atom = a2c[e];        // row 0 of [2, E]
    int cyc  = a2c[E + e];    // row 1 of [2, E]
    int code = x[atom];       // 0..21
    atomicAdd(&cnt[cyc * K_PAD + code], 1.0f);
}

// ---------------------------------------------------------------------------
// Kernel 3: out[M, 128] = Count[M, 24] @ Wpad[24, 128] via V_WMMA_F32_16X16X4_F32.
// Each block: 8 waves; wave w owns N columns [16w, 16w+16) and processes
// TILES_PER_BLOCK consecutive 16-row M-tiles, with the B fragments (6 x v2f)
// loaded once per wave and held in registers.
//
// Fragment layouts (wave32, f32):
//   A 16x4 : m = lane&15; lanes 0-15 hold K = {k0, k0+1}, lanes 16-31 K = {k0+2, k0+3}
//   B 4x16 : n = lane&15; same K split across lane halves
//   C/D 16x16: n = lane&15; VGPR r holds M = r (lanes 0-15) / r+8 (lanes 16-31)
//
// K-tail (k = 22, 23) handled branchlessly: clamp the emb row index to a valid
// address and v_cndmask the loaded value to zero -> no exec-mask branches.
// ---------------------------------------------------------------------------
__global__ __launch_bounds__(256) void gemm_wmma_kernel(
        const float* __restrict__ cnt,
        const float* __restrict__ emb,
        float* __restrict__ out) {
    const int mtile0 = blockIdx.x * TILES_PER_BLOCK;
    const int wave   = threadIdx.x >> 5;   // 0..7 -> N-tile
    const int lane   = threadIdx.x & 31;
    const int half   = lane >> 4;          // 0: K pair {k0,k0+1}; 1: {k0+2,k0+3}
    const int l      = lane & 15;          // m (for A) / n (for B, C/D)
    const int col    = wave * 16 + l;      // this lane's output column

    // ---- Load B fragments once (branchless zero-padding for k >= 22) ----
    v2f b[6];
    #pragma unroll
    for (int s = 0; s < 6; ++s) {
        const int k  = s * 4 + half * 2;
        const int k0 = (k     < K_DIM) ? k     : 0;   // clamped, always valid
        const int k1 = (k + 1 < K_DIM) ? k + 1 : 0;
        const float v0 = emb[k0 * N_DIM + col];
        const float v1 = emb[k1 * N_DIM + col];
        b[s].x = (k     < K_DIM) ? v0 : 0.0f;
        b[s].y = (k + 1 < K_DIM) ? v1 : 0.0f;
    }

    // ---- Loop over M tiles: A load -> 6 WMMA -> store ----
    #pragma unroll
    for (int t = 0; t < TILES_PER_BLOCK; ++t) {
        const int mtile = mtile0 + t;
        const float* arow =
            cnt + (size_t)(mtile * 16 + l) * K_PAD + half * 2;

        v8f c = {};
        #pragma unroll
        for (int s = 0; s < 6; ++s) {
            v2f a;
            a.x = arow[s * 4 + 0];
            a.y = arow[s * 4 + 1];
            // 8 args: (neg_a, A, neg_b, B, c_mod, C, reuse_a, reuse_b)
            c = __builtin_amdgcn_wmma_f32_16x16x4_f32(
                    false, a, false, b[s], (short)0, c, false, false);
        }

        float* o = out + (size_t)(mtile * 16) * N_DIM + col;
        #pragma unroll
        for (int r = 0; r < 8; ++r) {
            o[(size_t)(r + half * 8) * N_DIM] = c[r];
        }
    }
}

// ---------------------------------------------------------------------------
// Launcher. d_in[0] = x [100000] int, d_in[1] = atom_to_cycle [2,600000] int,
// d_in[2] = emb_weight [22,128] f32. d_out = [100000,128] f32.
// Workspace: count matrix, 100000*24*4 = 9.6 MB.
// ---------------------------------------------------------------------------
extern "C" void kernel_launch(void* const* d_in, const int* in_sizes, int n_in,
                              void* d_out, int out_size, void* d_ws, size_t ws_size,
                              hipStream_t stream) {
    const int*   x   = (const int*)d_in[0];
    const int*   a2c = (const int*)d_in[1];
    const float* emb = (const float*)d_in[2];
    float*       out = (float*)d_out;
    float*       cnt = (float*)d_ws;

    const int E    = in_sizes[1] / 2;          // 600000
    const int ncnt = NUM_SEG * K_PAD;          // 2.4M floats

    zero_counts_kernel<<<(ncnt + 255) / 256, 256, 0, stream>>>(cnt, ncnt);
    build_counts_kernel<<<(E + 255) / 256, 256, 0, stream>>>(x, a2c, cnt, E);
    gemm_wmma_kernel<<<NUM_BLOCKS, 256, 0, stream>>>(cnt, emb, out);
}